// GraphConvolution_31550829756520
// MI455X (gfx1250) — compile-verified
//
#include <hip/hip_runtime.h>
#include <hip/hip_bf16.h>

typedef __attribute__((ext_vector_type(2))) float v2f;
typedef __attribute__((ext_vector_type(8))) float v8f;

#define D_IN  256
#define D_OUT 128

#if defined(__gfx1250__) && __has_builtin(__builtin_amdgcn_global_load_async_to_lds_b128)
#define HAVE_ASYNC_LDS 1
#else
#define HAVE_ASYNC_LDS 0
#endif

// Types matching the builtin's signature: vector_size(16) int, AS1/AS3 ptrs.
typedef int v4i_vs __attribute__((__vector_size__(16)));
typedef __attribute__((address_space(1))) v4i_vs* gv4_ptr;
typedef __attribute__((address_space(3))) v4i_vs* lv4_ptr;

__device__ __forceinline__ void wait_asynccnt0() {
#if __has_builtin(__builtin_amdgcn_s_wait_asynccnt)
    __builtin_amdgcn_s_wait_asynccnt(0);
#else
    asm volatile("s_wait_asynccnt 0" ::: "memory");
#endif
}

// ---------------------------------------------------------------------------
// Kernel 1: support = feat @ W   (f32 WMMA 16x16x4)
// Block = 256 threads = 8 wave32s. Each wave computes a 16x128 strip of the
// output (8 tiles of 16x16 along D_OUT). Weight (256x128 f32 = 128 KB) is
// staged once per block into dynamic LDS (WGP has 320 KB) — via the CDNA5
// async global->LDS DMA path (ASYNCcnt-tracked, no VGPR data movement).
// ---------------------------------------------------------------------------
__global__ void gcn_gemm_wmma_f32(const float* __restrict__ feat,
                                  const float* __restrict__ weight,
                                  float* __restrict__ support,
                                  int n_nodes) {
    extern __shared__ float wlds[];   // D_IN * D_OUT floats = 128 KB

    const int tid  = threadIdx.x;
    const int lane = tid & 31;
    const int wave = tid >> 5;
    const int half = lane >> 4;   // 0: lanes 0-15, 1: lanes 16-31
    const int ln   = lane & 15;

#if HAVE_ASYNC_LDS
    // Async DMA: 8192 x b128 transfers, tracked by ASYNCcnt.
    {
        gv4_ptr gsrc = (gv4_ptr)weight;   // C-style cast: drop const, AS1
        lv4_ptr ldst = (lv4_ptr)wlds;     // AS3 (LDS)
        #pragma unroll 4
        for (int i = tid; i < (D_IN * D_OUT) / 4; i += 256) {
            __builtin_amdgcn_global_load_async_to_lds_b128(gsrc + i, ldst + i, 0, 0);
        }
        wait_asynccnt0();
    }
#else
    // Fallback: cooperative float4 copy through VGPRs.
    {
        const float4* wsrc = reinterpret_cast<const float4*>(weight);
        float4* wdst = reinterpret_cast<float4*>(wlds);
        #pragma unroll 4
        for (int i = tid; i < (D_IN * D_OUT) / 4; i += 256) {
            wdst[i] = wsrc[i];
        }
    }
#endif
    __syncthreads();

    const int rowBase = blockIdx.x * 128 + wave * 16;   // 16 rows per wave
    if (rowBase < n_nodes) {   // wave-uniform: EXEC stays all-ones for WMMA
        // n_nodes is a multiple of 16, so all 16 rows of this tile are valid.
        v8f c[8];
        #pragma unroll
        for (int n = 0; n < 8; ++n) c[n] = (v8f){0.f,0.f,0.f,0.f,0.f,0.f,0.f,0.f};

        // A fragment source: each lane streams one feat row.
        const float* arow = feat + (size_t)(rowBase + ln) * D_IN;

        for (int k0 = 0; k0 < D_IN; k0 += 4) {
            if ((k0 & 63) == 0) {
                // Pull the next chunk of this row toward the WGP caches.
                __builtin_prefetch(arow + k0 + 64, 0, 1);
            }
            // A 16x4 f32 fragment (ISA 7.12.2): lanes 0-15 hold K=k0,k0+1 in
            // v0,v1; lanes 16-31 hold K=k0+2,k0+3.
            v2f a;
            a.x = arow[k0 + half * 2 + 0];
            a.y = arow[k0 + half * 2 + 1];

            #pragma unroll
            for (int n = 0; n < 8; ++n) {
                // B 4x16 f32 fragment: rows (K) split across lane halves /
                // VGPRs, N striped across the 16 lanes of each half.
                v2f b;
                b.x = wlds[(k0 + half * 2 + 0) * D_OUT + n * 16 + ln];
                b.y = wlds[(k0 + half * 2 + 1) * D_OUT + n * 16 + ln];
                c[n] = __builtin_amdgcn_wmma_f32_16x16x4_f32(
                    /*neg_a=*/false, a, /*neg_b=*/false, b,
                    /*c_mod=*/(short)0, c[n],
                    /*reuse_a=*/false, /*reuse_b=*/false);
            }
        }

        // Store C/D: VGPR i -> (M=i, lanes 0-15) and (M=i+8, lanes 16-31).
        #pragma unroll
        for (int n = 0; n < 8; ++n) {
            #pragma unroll
            for (int i = 0; i < 8; ++i) {
                support[(size_t)(rowBase + i + half * 8) * D_OUT + n * 16 + ln] = c[n][i];
            }
        }
    }
}

// ---------------------------------------------------------------------------
// Kernel 2: out[r][c] = bias[c]   (initialize accumulation target)
// ---------------------------------------------------------------------------
__global__ void gcn_bias_init(const float* __restrict__ bias,
                              float* __restrict__ out,
                              int total) {
    int idx = blockIdx.x * blockDim.x + threadIdx.x;
    if (idx < total) {
        out[idx] = bias[idx & (D_OUT - 1)];
    }
}

// ---------------------------------------------------------------------------
// Kernel 3: edge-parallel SpMM scatter with f32 atomics.
// thread -> (edge, channel); gather is coalesced per edge, support/out are
// L2-resident (25.6 MB each vs 192 MB L2).
// ---------------------------------------------------------------------------
__global__ void gcn_spmm_scatter(const int* __restrict__ adj_row,
                                 const int* __restrict__ adj_col,
                                 const float* __restrict__ adj_vals,
                                 const float* __restrict__ support,
                                 float* __restrict__ out,
                                 int n_edges) {
    long long idx = (long long)blockIdx.x * blockDim.x + threadIdx.x;
    int e = (int)(idx >> 7);          // D_OUT == 128 channels per edge
    int ch = (int)(idx & (D_OUT - 1));
    if (e < n_edges) {
        int r = adj_row[e];
        int c = adj_col[e];
        float v = adj_vals[e] * support[(size_t)c * D_OUT + ch];
        atomicAdd(&out[(size_t)r * D_OUT + ch], v);
    }
}

extern "C" void kernel_launch(void* const* d_in, const int* in_sizes, int n_in,
                              void* d_out, int out_size, void* d_ws, size_t ws_size,
                              hipStream_t stream) {
    const float* feat     = (const float*)d_in[0];
    const int*   adj_row  = (const int*)d_in[1];
    const int*   adj_col  = (const int*)d_in[2];
    const float* adj_vals = (const float*)d_in[3];
    const float* weight   = (const float*)d_in[4];
    const float* bias     = (const float*)d_in[5];
    float* out     = (float*)d_out;
    float* support = (float*)d_ws;          // n_nodes * D_OUT floats (25.6 MB)

    const int n_nodes = in_sizes[0] / D_IN; // 50000
    const int n_edges = in_sizes[1];        // 800000

    // 1) Dense GEMM: support = feat @ W
    {
        dim3 block(256);
        dim3 grid((n_nodes + 127) / 128);
        size_t lds_bytes = (size_t)D_IN * D_OUT * sizeof(float); // 128 KB
        gcn_gemm_wmma_f32<<<grid, block, lds_bytes, stream>>>(feat, weight, support, n_nodes);
    }

    // 2) out = broadcast(bias)
    {
        int total = n_nodes * D_OUT;
        gcn_bias_init<<<(total + 255) / 256, 256, 0, stream>>>(bias, out, total);
    }

    // 3) Scatter-accumulate over edges
    {
        long long total = (long long)n_edges * D_OUT;
        int blocks = (int)((total + 255) / 256);
        gcn_spmm_scatter<<<blocks, 256, 0, stream>>>(adj_row, adj_col, adj_vals,
                                                     support, out, n_edges);
    }
}